// STGCN_395136991874
// MI455X (gfx1250) — compile-verified
//
#include <hip/hip_runtime.h>

// Problem constants (from reference): B=32, Z=4000, IN=32, H=64, E=64000
#define B_   32
#define Z_   4000
#define IN_  32
#define H_   64
#define E_   64000
#define R_   (B_ * Z_)   // 128000 rows of x / elements of out

typedef __attribute__((ext_vector_type(2))) float v2f;
typedef __attribute__((ext_vector_type(8))) float v8f;

// ---------------------------------------------------------------------------
// K1: deg[z] = 1 (self loop)
__global__ __launch_bounds__(256) void gcn_deg_init(unsigned* __restrict__ deg) {
    int z = blockIdx.x * blockDim.x + threadIdx.x;
    if (z < Z_) deg[z] = 1u;
}

// K2: deg[dst[e]] += 1
__global__ __launch_bounds__(256) void gcn_deg_count(const int* __restrict__ ei,
                                                     unsigned* __restrict__ deg) {
    int e = blockIdx.x * blockDim.x + threadIdx.x;
    if (e < E_) atomicAdd(&deg[ei[E_ + e]], 1u);
}

// K3: dinv[z] = deg^-1/2  (deg >= 1 always due to self loop)
__global__ __launch_bounds__(256) void gcn_dinv(const unsigned* __restrict__ deg,
                                                float* __restrict__ dinv) {
    int z = blockIdx.x * blockDim.x + threadIdx.x;
    if (z < Z_) {
        unsigned d = deg[z];
        dinv[z] = (d > 0u) ? rsqrtf((float)d) : 0.0f;
    }
}

// K4: fold the FC head through the linear map:
//     v[i] = sum_h W[i,h]*fcW[h] ;  c = sum_h b[h]*fcW[h] + fcb
__global__ __launch_bounds__(64) void gcn_prep(const float* __restrict__ W,
                                               const float* __restrict__ bias,
                                               const float* __restrict__ fcW,
                                               const float* __restrict__ fcb,
                                               float* __restrict__ v,
                                               float* __restrict__ c) {
    int t = threadIdx.x;
    if (t < IN_) {
        float s = 0.0f;
        #pragma unroll
        for (int h = 0; h < H_; ++h) s += W[t * H_ + h] * fcW[h];
        v[t] = s;
    } else if (t == IN_) {
        float s = fcb[0];
        #pragma unroll
        for (int h = 0; h < H_; ++h) s += bias[h] * fcW[h];
        c[0] = s;
    }
}

// ---------------------------------------------------------------------------
// K5: y[r] = x[r,:] . v   via V_WMMA_F32_16X16X4_F32 (full f32 precision).
// One wave owns 16 consecutive rows; 8 chained WMMAs cover K=32.
// A 16x4 f32 layout (ISA 7.12.2): lane = row (lane&15), lane-half selects K pair,
//   VGPR0/VGPR1 = consecutive K. B 4x16 mirrors it per-lane; v is replicated
//   across all 16 columns, so every column of D holds the same dot product.
// D layout: acc[j] = row j (lanes 0-15) / row j+8 (lanes 16-31).
__global__ __launch_bounds__(256) void gcn_xv_wmma(const float* __restrict__ x,
                                                   const float* __restrict__ v,
                                                   float* __restrict__ y) {
    const int lane = threadIdx.x & 31;
    const int wave = threadIdx.x >> 5;
    const int row_base = blockIdx.x * 128 + wave * 16;   // 16 rows per wave
    const int half = lane >> 4;                          // 0 | 1
    const int m    = lane & 15;                          // row within tile

    const float* xrow = x + (size_t)(row_base + m) * IN_;

    v8f acc = {};
    #pragma unroll
    for (int ks = 0; ks < 8; ++ks) {
        const int k0 = ks * 4 + half * 2;
        v2f a;  a.x  = xrow[k0];  a.y  = xrow[k0 + 1];
        v2f bm; bm.x = v[k0];     bm.y = v[k0 + 1];
        // 8 args: (neg_a, A, neg_b, B, c_mod, C, reuse_a, reuse_b)
        acc = __builtin_amdgcn_wmma_f32_16x16x4_f32(
                  false, a, false, bm, (short)0, acc, false, false);
    }

    // All 16 columns identical: lanes 0 and 16 write the 16 row results.
    if (m == 0) {
        float* yo = y + row_base + half * 8;
        #pragma unroll
        for (int j = 0; j < 8; ++j) yo[j] = acc[j];
    }
}

// ---------------------------------------------------------------------------
// K6: out[r] = dinv[z]^2 * y[r] + c    (self-loop message + folded bias)
__global__ __launch_bounds__(256) void gcn_self(const float* __restrict__ y,
                                                const float* __restrict__ dinv,
                                                const float* __restrict__ c,
                                                float* __restrict__ out) {
    int r = blockIdx.x * blockDim.x + threadIdx.x;
    if (r < R_) {
        int z = r % Z_;
        float di = dinv[z];
        out[r] = di * di * y[r] + c[0];
    }
}

// K7: out[b, dst] += dinv[src]*dinv[dst] * y[b, src]  for every edge.
// y and out are 512 KB each -> fully L2-resident on a 192 MB L2; hardware
// f32 atomics (global_atomic_add_f32) via unsafeAtomicAdd.
__global__ __launch_bounds__(256) void gcn_scatter(const int* __restrict__ ei,
                                                   const float* __restrict__ dinv,
                                                   const float* __restrict__ y,
                                                   float* __restrict__ out) {
    int e = blockIdx.x * blockDim.x + threadIdx.x;
    if (e < E_) {
        int s = ei[e];
        int d = ei[E_ + e];
        float w = dinv[s] * dinv[d];
        #pragma unroll 4
        for (int bb = 0; bb < B_; ++bb) {
            unsafeAtomicAdd(&out[bb * Z_ + d], w * y[bb * Z_ + s]);
        }
    }
}

// ---------------------------------------------------------------------------
extern "C" void kernel_launch(void* const* d_in, const int* in_sizes, int n_in,
                              void* d_out, int out_size, void* d_ws, size_t ws_size,
                              hipStream_t stream) {
    const float* x   = (const float*)d_in[0];   // [B, Z, IN]
    const int*   ei  = (const int*)  d_in[1];   // [2, E]
    const float* W   = (const float*)d_in[2];   // [IN, H]
    const float* bia = (const float*)d_in[3];   // [H]
    const float* fcW = (const float*)d_in[4];   // [H, 1]
    const float* fcb = (const float*)d_in[5];   // [1]
    float*       out = (float*)d_out;           // [B, Z]

    // Workspace layout (floats): deg | dinv | v | c | y   (~546 KB total)
    unsigned* deg  = (unsigned*)d_ws;                 // 4000 u32   @ 0
    float*    dinv = (float*)d_ws + 4096;             // 4000 f32
    float*    v    = (float*)d_ws + 8192;             // 32 f32
    float*    c    = (float*)d_ws + 8256;             // 1 f32
    float*    y    = (float*)d_ws + 8448;             // 128000 f32

    gcn_deg_init <<<(Z_ + 255) / 256, 256, 0, stream>>>(deg);
    gcn_deg_count<<<(E_ + 255) / 256, 256, 0, stream>>>(ei, deg);
    gcn_dinv     <<<(Z_ + 255) / 256, 256, 0, stream>>>(deg, dinv);
    gcn_prep     <<<1, 64, 0, stream>>>(W, bia, fcW, fcb, v, c);
    gcn_xv_wmma  <<<R_ / 128, 256, 0, stream>>>(x, v, y);
    gcn_self     <<<(R_ + 255) / 256, 256, 0, stream>>>(y, dinv, c, out);
    gcn_scatter  <<<(E_ + 255) / 256, 256, 0, stream>>>(ei, dinv, y, out);
}